// GNNRegressor_84825604096523
// MI455X (gfx1250) — compile-verified
//
#include <hip/hip_runtime.h>
#include <hip/hip_bf16.h>

typedef __attribute__((ext_vector_type(2))) float v2f;
typedef __attribute__((ext_vector_type(8))) float v8f;

#define HID 64

__device__ __forceinline__ void atomicAddF(float* p, float v) {
    __hip_atomic_fetch_add(p, v, __ATOMIC_RELAXED, __HIP_MEMORY_SCOPE_AGENT);
}

// ---------------- degree / normalization ----------------

__global__ void init_deg_gsum(float* __restrict__ deg, float* __restrict__ gsum, int N) {
    int i = blockIdx.x * blockDim.x + threadIdx.x;
    if (i < N) deg[i] = 1.0f;          // self-loop contributes 1 to every degree
    if (i < HID) gsum[i] = 0.0f;
}

__global__ void deg_edges(const int* __restrict__ dst, float* __restrict__ deg, int E) {
    int e = blockIdx.x * blockDim.x + threadIdx.x;
    if (e < E) atomicAddF(&deg[dst[e]], 1.0f);
}

__global__ void deg_to_dinv(float* __restrict__ deg, int N) {
    int i = blockIdx.x * blockDim.x + threadIdx.x;
    if (i < N) deg[i] = rsqrtf(deg[i]);   // deg >= 1 always (self loops)
}

// ---------------- fp32 WMMA GEMM: out[M,64] = (relu?)A[M,K] @ W[K,64] ----------------
// One wave per 16x16 output tile; V_WMMA_F32_16X16X4_F32, K stepped by 4.

template <int K, bool RELU_IN>
__global__ void gemm_wmma_f32(const float* __restrict__ A, const float* __restrict__ W,
                              float* __restrict__ out, int M) {
    int gtid = blockIdx.x * blockDim.x + threadIdx.x;
    int wave = gtid >> 5;                 // uniform within wave
    int lane = threadIdx.x & 31;
    int mtiles = (M + 15) >> 4;
    int mt = wave >> 2;                   // 4 column tiles (N=64)
    int nt = wave & 3;
    if (mt >= mtiles) return;             // wave-uniform: EXEC all-ones below

    int mrow  = mt * 16 + (lane & 15);
    int mload = mrow < M ? mrow : (M - 1);   // clamp loads (keep EXEC full)
    int n0    = nt * 16;
    int khalf = (lane >> 4) << 1;         // 0 or 2
    int ncol  = n0 + (lane & 15);

    v8f c = {};
    #pragma unroll
    for (int k0 = 0; k0 < K; k0 += 4) {
        float a0 = A[mload * K + k0 + khalf + 0];
        float a1 = A[mload * K + k0 + khalf + 1];
        if (RELU_IN) { a0 = fmaxf(a0, 0.0f); a1 = fmaxf(a1, 0.0f); }
        v2f a; a[0] = a0; a[1] = a1;
        v2f b;
        b[0] = W[(k0 + khalf + 0) * HID + ncol];
        b[1] = W[(k0 + khalf + 1) * HID + ncol];
        // D = A*B + C : v_wmma_f32_16x16x4_f32
        c = __builtin_amdgcn_wmma_f32_16x16x4_f32(false, a, false, b, (short)0, c,
                                                  false, false);
    }

    int row_base = mt * 16 + ((lane >> 4) << 3);   // +8 for upper half-wave
    float* outp  = out + (size_t)row_base * HID + ncol;
    if (mt * 16 + 16 <= M) {
        // wave-uniform full-tile fast path: straight-line stores, no EXEC churn
        #pragma unroll
        for (int r = 0; r < 8; ++r) outp[r * HID] = c[r];
    } else {
        #pragma unroll
        for (int r = 0; r < 8; ++r) {
            if (row_base + r < M) outp[r * HID] = c[r];
        }
    }
}

// ---------------- aggregation ----------------

// self-loop + bias: out[i,f] = b[f] + h[i,f] * dinv[i]^2   (norm of self edge = 1/deg)
__global__ void agg_init(const float* __restrict__ h, const float* __restrict__ dinv,
                         const float* __restrict__ bias, float* __restrict__ out, int N) {
    int g = blockIdx.x * blockDim.x + threadIdx.x;
    if (g >= N * HID) return;
    int i = g >> 6, f = g & (HID - 1);
    float di = dinv[i];
    out[g] = bias[f] + h[g] * di * di;
}

// 16 lanes per edge, float4 per lane: coalesced 256B gather of h[src], L2 atomic scatter to out[dst]
__global__ void agg_edges(const int* __restrict__ src, const int* __restrict__ dst,
                          const float* __restrict__ dinv, const float* __restrict__ h,
                          float* __restrict__ out, int E) {
    int g = blockIdx.x * blockDim.x + threadIdx.x;
    int e = g >> 4;
    if (e >= E) return;
    int c = (g & 15) << 2;
    int s = src[e], d = dst[e];
    float w = dinv[s] * dinv[d];
    const float4 v = *(const float4*)(h + (size_t)s * HID + c);
    float* o = out + (size_t)d * HID + c;
    atomicAddF(o + 0, v.x * w);
    atomicAddF(o + 1, v.y * w);
    atomicAddF(o + 2, v.z * w);
    atomicAddF(o + 3, v.w * w);
}

// ---------------- readout ----------------

__global__ void readout_partial(const float* __restrict__ a2, float* __restrict__ gsum, int N) {
    int f  = threadIdx.x & (HID - 1);
    int i0 = blockIdx.x * (blockDim.x / HID) + (threadIdx.x >> 6);
    int stride = gridDim.x * (blockDim.x / HID);
    float acc = 0.0f;
    for (int i = i0; i < N; i += stride)
        acc += fmaxf(a2[(size_t)i * HID + f], 0.0f);   // relu fused
    atomicAddF(&gsum[f], acc);
}

__global__ void final_fc(const float* __restrict__ gsum, const float* __restrict__ Wfc,
                         const float* __restrict__ bfc, float* __restrict__ out, float invN) {
    int j = threadIdx.x;
    if (j >= 16) return;
    float acc = bfc[j];
    #pragma unroll
    for (int f = 0; f < HID; ++f)
        acc += (gsum[f] * invN) * Wfc[f * 16 + j];
    out[j] = acc;
}

// ---------------- launcher ----------------

extern "C" void kernel_launch(void* const* d_in, const int* in_sizes, int n_in,
                              void* d_out, int out_size, void* d_ws, size_t ws_size,
                              hipStream_t stream) {
    const float* x   = (const float*)d_in[0];
    const int*   src = (const int*)d_in[1];
    const int*   dst = (const int*)d_in[2];
    const float* W1  = (const float*)d_in[3];
    const float* b1  = (const float*)d_in[4];
    const float* W2  = (const float*)d_in[5];
    const float* b2  = (const float*)d_in[6];
    const float* Wfc = (const float*)d_in[7];
    const float* bfc = (const float*)d_in[8];
    float* out = (float*)d_out;

    const int N = in_sizes[0] / 32;   // in_dim = 32
    const int E = in_sizes[1];

    float* ws   = (float*)d_ws;
    float* dinv = ws;                                  // N floats (deg -> rsqrt in place)
    float* gsum = ws + N;                              // 64 floats
    size_t base = ((size_t)N + HID + 255) & ~(size_t)255;
    float* bufA = ws + base;                           // N*64
    float* bufB = bufA + (size_t)N * HID;              // N*64

    const int B = 256;

    // normalization (graph-only, reused by both layers)
    init_deg_gsum<<<(N + B - 1) / B, B, 0, stream>>>(dinv, gsum, N);
    deg_edges<<<(E + B - 1) / B, B, 0, stream>>>(dst, dinv, E);
    deg_to_dinv<<<(N + B - 1) / B, B, 0, stream>>>(dinv, N);

    // layer 1: h1 = x @ W1 ; a1 = scatter(h1) + b1
    {
        int mtiles = (N + 15) >> 4;
        int waves  = mtiles * 4;
        int blocks = (waves * 32 + B - 1) / B;
        gemm_wmma_f32<32, false><<<blocks, B, 0, stream>>>(x, W1, bufA, N);
    }
    agg_init<<<((size_t)N * HID + B - 1) / B, B, 0, stream>>>(bufA, dinv, b1, bufB, N);
    agg_edges<<<((size_t)E * 16 + B - 1) / B, B, 0, stream>>>(src, dst, dinv, bufA, bufB, E);

    // layer 2: h2 = relu(a1) @ W2 ; a2 = scatter(h2) + b2   (relu fused into GEMM A-load)
    {
        int mtiles = (N + 15) >> 4;
        int waves  = mtiles * 4;
        int blocks = (waves * 32 + B - 1) / B;
        gemm_wmma_f32<64, true><<<blocks, B, 0, stream>>>(bufB, W2, bufA, N);
    }
    agg_init<<<((size_t)N * HID + B - 1) / B, B, 0, stream>>>(bufA, dinv, b2, bufB, N);
    agg_edges<<<((size_t)E * 16 + B - 1) / B, B, 0, stream>>>(src, dst, dinv, bufA, bufB, E);

    // readout: g = mean(relu(a2)) ; out = g @ Wfc + bfc
    readout_partial<<<512, B, 0, stream>>>(bufB, gsum, N);
    final_fc<<<1, 32, 0, stream>>>(gsum, Wfc, bfc, out, 1.0f / (float)N);
}